// GCN_15960098472722
// MI455X (gfx1250) — compile-verified
//
#include <hip/hip_runtime.h>
#include <math.h>

typedef __attribute__((ext_vector_type(2))) float v2f;
typedef __attribute__((ext_vector_type(8))) float v8f;

// ---------------- degree / norm ----------------
__global__ void deg_init_kernel(float* deg, int n) {
    int i = blockIdx.x * blockDim.x + threadIdx.x;
    if (i < n) deg[i] = 1.0f;  // self-loop contribution
}

__global__ void deg_count_kernel(const long long* __restrict__ dst, float* deg, int e) {
    int i = blockIdx.x * blockDim.x + threadIdx.x;
    if (i < e) unsafeAtomicAdd(&deg[(int)dst[i]], 1.0f);
}

__global__ void dinv_kernel(float* deg, int n) {
    int i = blockIdx.x * blockDim.x + threadIdx.x;
    if (i < n) {
        float d = deg[i];
        deg[i] = (d > 0.0f) ? rsqrtf(d) : 0.0f;
    }
}

// ---------------- layer-1 GEMM: h1 = x @ W1 (128 -> 16), fp32 WMMA ----------------
// one wave per 16-row tile; n assumed divisible by 16 (100000 = 6250*16)
__global__ void __launch_bounds__(32) gemm1_kernel(const float* __restrict__ x,
                                                   const float* __restrict__ W1,
                                                   float* __restrict__ h1) {
    __shared__ float tile[16 * 128];   // 8 KB per (single-wave) block
    const int lane = threadIdx.x;      // 0..31
    const int row0 = blockIdx.x * 16;

    // cooperative coalesced load of the 16x128 fp32 tile (float4 = b128)
    const float4* src = (const float4*)(x + (size_t)row0 * 128);
    float4* dstp = (float4*)tile;
#pragma unroll
    for (int i = 0; i < 16; ++i)
        dstp[i * 32 + lane] = src[i * 32 + lane];
    __syncthreads();  // single-wave workgroup: orders LDS writes vs reads

    const int nn = lane & 15;           // N (and M) index within tile
    const int kb = (lane >> 4) * 2;     // lane-half K offset for A/B operands

    // preload all of W1 as B operands (32 chunks of K=4)
    v2f B[32];
#pragma unroll
    for (int k = 0; k < 32; ++k) {
        int kk = k * 4 + kb;
        B[k].x = W1[(size_t)kk * 16 + nn];
        B[k].y = W1[(size_t)(kk + 1) * 16 + nn];
    }

    v8f c = {};
#pragma unroll
    for (int k = 0; k < 32; ++k) {
        int kk = k * 4 + kb;
        v2f a;
        a.x = tile[nn * 128 + kk];
        a.y = tile[nn * 128 + kk + 1];
        c = __builtin_amdgcn_wmma_f32_16x16x4_f32(false, a, false, B[k],
                                                  (short)0, c, false, false);
    }

    // store D: VGPR r holds M = r + 8*(lane>=16), N = lane%16
    const int mh = (lane >> 4) * 8;
#pragma unroll
    for (int r = 0; r < 8; ++r)
        h1[(size_t)(row0 + mh + r) * 16 + nn] = c[r];
}

// ---------------- aggregation ----------------
// self-loop term: out[i,f] = h[i,f] * dinv[i]^2
__global__ void agg_self_kernel(const float* __restrict__ dinv,
                                const float* __restrict__ h,
                                float* __restrict__ out, long long work) {
    long long t = (long long)blockIdx.x * blockDim.x + threadIdx.x;
    if (t >= work) return;
    int i = (int)(t >> 4);
    float di = dinv[i];
    out[t] = h[t] * di * di;
}

// edge term: out[dst,f] += h[src,f] * dinv[src]*dinv[dst]
__global__ void agg_edges_kernel(const long long* __restrict__ srcI,
                                 const long long* __restrict__ dstI,
                                 const float* __restrict__ dinv,
                                 const float* __restrict__ h,
                                 float* __restrict__ out, long long work) {
    long long t = (long long)blockIdx.x * blockDim.x + threadIdx.x;
    if (t >= work) return;
    int e = (int)(t >> 4);
    int f = (int)(t & 15);
    int s = (int)srcI[e];
    int d = (int)dstI[e];
    float norm = dinv[s] * dinv[d];
    unsafeAtomicAdd(&out[(size_t)d * 16 + f], h[(size_t)s * 16 + f] * norm);
}

// ---------------- layer-2 GEMM: h2 = relu(agg1 + b1) @ W2 (16 -> 10, padded to 16) ----------------
__global__ void __launch_bounds__(32) gemm2_kernel(const float* __restrict__ agg1,
                                                   const float* __restrict__ b1,
                                                   const float* __restrict__ W2,
                                                   float* __restrict__ h2) {
    const int lane = threadIdx.x;
    const int row0 = blockIdx.x * 16;
    const int nn = lane & 15;
    const int kb = (lane >> 4) * 2;

    v2f B[4];
#pragma unroll
    for (int k = 0; k < 4; ++k) {
        int kk = k * 4 + kb;
        B[k].x = (nn < 10) ? W2[(size_t)kk * 10 + nn] : 0.0f;
        B[k].y = (nn < 10) ? W2[(size_t)(kk + 1) * 10 + nn] : 0.0f;
    }

    v8f c = {};
#pragma unroll
    for (int k = 0; k < 4; ++k) {
        int kk = k * 4 + kb;
        const float* rowp = agg1 + (size_t)(row0 + nn) * 16;
        v2f a;
        a.x = fmaxf(rowp[kk] + b1[kk], 0.0f);        // fused bias + relu
        a.y = fmaxf(rowp[kk + 1] + b1[kk + 1], 0.0f);
        c = __builtin_amdgcn_wmma_f32_16x16x4_f32(false, a, false, B[k],
                                                  (short)0, c, false, false);
    }

    const int mh = (lane >> 4) * 8;
#pragma unroll
    for (int r = 0; r < 8; ++r)
        h2[(size_t)(row0 + mh + r) * 16 + nn] = c[r];
}

// ---------------- bias + log_softmax over 10 classes ----------------
__global__ void logsoftmax_kernel(const float* __restrict__ agg2,
                                  const float* __restrict__ b2,
                                  float* __restrict__ out, int n) {
    int i = blockIdx.x * blockDim.x + threadIdx.x;
    if (i >= n) return;
    float v[10];
    float mx = -3.4e38f;
#pragma unroll
    for (int cix = 0; cix < 10; ++cix) {
        v[cix] = agg2[(size_t)i * 16 + cix] + b2[cix];
        mx = fmaxf(mx, v[cix]);
    }
    float s = 0.0f;
#pragma unroll
    for (int cix = 0; cix < 10; ++cix) s += expf(v[cix] - mx);
    float lse = mx + logf(s);
#pragma unroll
    for (int cix = 0; cix < 10; ++cix)
        out[(size_t)i * 10 + cix] = v[cix] - lse;
}

extern "C" void kernel_launch(void* const* d_in, const int* in_sizes, int n_in,
                              void* d_out, int out_size, void* d_ws, size_t ws_size,
                              hipStream_t stream) {
    const float*     x    = (const float*)d_in[0];
    const long long* edge = (const long long*)d_in[1];
    const float*     W1   = (const float*)d_in[2];
    const float*     b1   = (const float*)d_in[3];
    const float*     W2   = (const float*)d_in[4];
    const float*     b2   = (const float*)d_in[5];

    const int n = in_sizes[0] / 128;   // 100000
    const int e = in_sizes[1] / 2;     // 3200000
    const long long* srcI = edge;       // edge_index[0]
    const long long* dstI = edge + e;   // edge_index[1]

    float* dinv = (float*)d_ws;                 // n floats
    float* bufA = dinv + n;                     // n*16 floats (h1, later h2)
    float* bufB = bufA + (size_t)n * 16;        // n*16 floats (agg1, later agg2)

    const int tiles = n / 16;                   // 6250, exact
    const long long nodeW = (long long)n * 16;
    const long long edgeW = (long long)e * 16;
    const int tB = 256;

    // degrees -> dinv (in place)
    deg_init_kernel<<<(n + tB - 1) / tB, tB, 0, stream>>>(dinv, n);
    deg_count_kernel<<<(e + tB - 1) / tB, tB, 0, stream>>>(dstI, dinv, e);
    dinv_kernel<<<(n + tB - 1) / tB, tB, 0, stream>>>(dinv, n);

    // layer 1: h1 = x @ W1 ; agg1 = norm-aggregate(h1)
    gemm1_kernel<<<tiles, 32, 0, stream>>>(x, W1, bufA);
    agg_self_kernel<<<(int)((nodeW + tB - 1) / tB), tB, 0, stream>>>(dinv, bufA, bufB, nodeW);
    agg_edges_kernel<<<(int)((edgeW + tB - 1) / tB), tB, 0, stream>>>(srcI, dstI, dinv, bufA, bufB, edgeW);

    // layer 2: h2 = relu(agg1 + b1) @ W2 ; agg2 = norm-aggregate(h2)
    gemm2_kernel<<<tiles, 32, 0, stream>>>(bufB, b1, W2, bufA);
    agg_self_kernel<<<(int)((nodeW + tB - 1) / tB), tB, 0, stream>>>(dinv, bufA, bufB, nodeW);
    agg_edges_kernel<<<(int)((edgeW + tB - 1) / tB), tB, 0, stream>>>(srcI, dstI, dinv, bufA, bufB, edgeW);

    // bias + log_softmax -> d_out
    logsoftmax_kernel<<<(n + tB - 1) / tB, tB, 0, stream>>>(bufB, b2, (float*)d_out, n);
}